// NuclearLossFunc_58420145160869
// MI455X (gfx1250) — compile-verified
//
#include <hip/hip_runtime.h>

// NuclearLossFunc on MI455X (gfx1250):
//   loss = sum(x^2) / (B*C),  x : 32x64x256x256 fp32  (512 MiB, B*C = 2048)
// Pure HBM-bound streaming reduction: floor = 512MiB / 23.3 TB/s ~= 23 us.
// Strategy: NT-hinted b128 loads (no reuse, working set > 192MB L2), fp32 FMA
// accumulation, and V_WMMA_F32_16X16X4_F32 (exact fp32 WMMA) as the 32-lane
// cross-lane reducer. Two-pass deterministic tree reduction, no atomics.

typedef float v2f __attribute__((ext_vector_type(2)));
typedef float v4f __attribute__((ext_vector_type(4)));
typedef float v8f __attribute__((ext_vector_type(8)));

#define BLOCK 256   // 8 waves (wave32)
#define GRID  2048  // 16K waves chip-wide; partials = 8 KB in d_ws

// Reduce 32 per-lane fp32 values to one scalar (broadcast to all lanes) using
// V_WMMA_F32_16X16X4_F32.
// A-matrix 16x4 fp32 layout (ISA 7.12.2): lane L<16 holds A[L][0] in VGPR0,
// A[L][1] in VGPR1; lane L>=16 holds A[L-16][2], A[L-16][3]. So a = {s, 0}
// places all 32 lane partials into A. With B = ones(4x16):
//   D[m][n] = s_m + s_{m+16}  (independent of n).
// D layout: VGPR v holds D[v][*] in lanes 0-15 and D[v+8][*] in lanes 16-31,
// so sum of the 8 D VGPRs gives rows 0..7 (lanes 0-15) / rows 8..15 (lanes
// 16-31); one shfl_xor(16) completes the full 32-lane sum.
// EXEC must be all-ones at the WMMA: callers keep whole waves active.
__device__ __forceinline__ float wave_reduce_wmma(float s) {
    v2f a; a[0] = s;    a[1] = 0.0f;
    v2f b; b[0] = 1.0f; b[1] = 1.0f;
    v8f c = {};
    v8f d = __builtin_amdgcn_wmma_f32_16x16x4_f32(
        /*neg_a=*/false, a, /*neg_b=*/false, b,
        /*c_mod=*/(short)0, c, /*reuse_a=*/false, /*reuse_b=*/false);
    float r = ((d[0] + d[1]) + (d[2] + d[3])) + ((d[4] + d[5]) + (d[6] + d[7]));
    r += __shfl_xor(r, 16, 32);
    return r;
}

// Block-level reduction: WMMA per wave, 8 wave totals through LDS, WMMA again
// in wave 0. Result valid in wave 0 (thread 0 uses it).
__device__ __forceinline__ float block_reduce(float s) {
    __shared__ float lds[BLOCK / 32];
    const int lane = threadIdx.x & 31;
    const int wave = threadIdx.x >> 5;

    float w = wave_reduce_wmma(s);
    if (lane == 0) lds[wave] = w;
    __syncthreads();

    float total = 0.0f;
    if (wave == 0) {  // wave-uniform branch: EXEC all-ones inside
        float v = (lane < (BLOCK / 32)) ? lds[lane] : 0.0f;  // cndmask, no divergence
        total = wave_reduce_wmma(v);
    }
    return total;
}

__global__ void __launch_bounds__(BLOCK)
sumsq_partial_kernel(const v4f* __restrict__ in, float* __restrict__ partial,
                     unsigned long long n4) {
    const unsigned long long tid =
        (unsigned long long)blockIdx.x * BLOCK + threadIdx.x;
    const unsigned long long stride = (unsigned long long)GRID * BLOCK;

    // n4 (33,554,432) is an exact multiple of stride (524,288): every lane runs
    // the same 64 iterations -> no divergence, EXEC stays all-ones for WMMA.
    float a0 = 0.0f, a1 = 0.0f, a2 = 0.0f, a3 = 0.0f;
    for (unsigned long long i = tid; i < n4; i += stride) {
        v4f v = __builtin_nontemporal_load(&in[i]);  // global_load_b128, TH=NT
        a0 = fmaf(v[0], v[0], a0);
        a1 = fmaf(v[1], v[1], a1);
        a2 = fmaf(v[2], v[2], a2);
        a3 = fmaf(v[3], v[3], a3);
    }
    float s = (a0 + a1) + (a2 + a3);

    float t = block_reduce(s);
    if (threadIdx.x == 0) partial[blockIdx.x] = t;
}

__global__ void __launch_bounds__(BLOCK)
final_reduce_kernel(const float* __restrict__ partial, float* __restrict__ out,
                    int n, float inv_bc) {
    // n == GRID (2048): 8 uniform iterations per thread, no divergence.
    float s = 0.0f;
    for (int i = threadIdx.x; i < n; i += BLOCK) s += partial[i];

    float t = block_reduce(s);
    if (threadIdx.x == 0) out[0] = t * inv_bc;
}

extern "C" void kernel_launch(void* const* d_in, const int* in_sizes, int n_in,
                              void* d_out, int out_size, void* d_ws, size_t ws_size,
                              hipStream_t stream) {
    (void)n_in; (void)out_size; (void)ws_size;

    const v4f* in = (const v4f*)d_in[0];
    const unsigned long long n  = (unsigned long long)in_sizes[0];  // 134,217,728
    const unsigned long long n4 = n >> 2;                           // float4 count

    float* partial = (float*)d_ws;  // GRID floats = 8 KB, fully rewritten each call

    // Reference divisor: B*C = 32*64 = 2048.
    const float inv_bc = 1.0f / 2048.0f;

    sumsq_partial_kernel<<<GRID, BLOCK, 0, stream>>>(in, partial, n4);
    final_reduce_kernel<<<1, BLOCK, 0, stream>>>(partial, (float*)d_out, GRID, inv_bc);
}